// FlashQwenAttention_4415226380861
// MI455X (gfx1250) — compile-verified
//
#include <hip/hip_runtime.h>

#define H     32
#define D     128
#define BS    16
#define BATCH 8
#define KVLEN 2048
#define HID   4096      // H*D
#define QKV3  12288     // 3*H*D
#define SCALE 0.08838834764831845f   // 1/sqrt(128)

typedef float v2f __attribute__((ext_vector_type(2)));
typedef float v8f __attribute__((ext_vector_type(8)));

// ---------------------------------------------------------------------------
// GEMM: out[m, n] = sum_k A[m, k] * W[n, k] (+ bias[n]), m in [0,8), K = HID.
// One wave per 16-wide N tile, M padded 8->16 with V_WMMA_F32_16X16X4_F32.
//
// K-mapping trick: per 16-K step, lanes 0-15 carry k{0..7} and lanes 16-31
// carry k{8..15}; wmma_j consumes pair {2j,2j+1} of each half. A and B use
// the same mapping, so the accumulation over all 16 k's is exact, and each
// lane's 8 floats are contiguous -> two global_load_b128 per operand.
//
// A rows 8-15 feed only C rows M=8-15, which are discarded at store time, so
// invalid rows alias row&7 (valid address, don't-care data) -- no branch.
// ---------------------------------------------------------------------------
__global__ __launch_bounds__(256) void gemm_aw_t(
    const float* __restrict__ A,     // 8 x K
    const float* __restrict__ W,     // N x K (row-major; computes A @ W^T)
    const float* __restrict__ bias,  // N or nullptr
    float* __restrict__ out,         // 8 x N
    int N, int K)
{
    const int lane = threadIdx.x & 31;
    const int wave = threadIdx.x >> 5;
    const int tile = blockIdx.x * 8 + wave;
    const int n0   = tile * 16;
    if (n0 >= N) return;                          // wave-uniform

    const int row  = lane & 15;
    const int koff = (lane >> 4) * 8;             // lanes 0-15 -> k 0..7, 16-31 -> 8..15
    const float* __restrict__ arow = A + (size_t)(row & 7) * K;   // rows>=8: don't-care alias
    const float* __restrict__ brow = W + (size_t)(n0 + row) * K;

    v8f c = {};
    #pragma unroll 2
    for (int k = 0; k < K; k += 16) {
        const float4 alo = *(const float4*)(arow + k + koff);
        const float4 ahi = *(const float4*)(arow + k + koff + 4);
        const float4 blo = *(const float4*)(brow + k + koff);
        const float4 bhi = *(const float4*)(brow + k + koff + 4);
        v2f a0 = {alo.x, alo.y}, a1 = {alo.z, alo.w};
        v2f a2 = {ahi.x, ahi.y}, a3 = {ahi.z, ahi.w};
        v2f b0 = {blo.x, blo.y}, b1 = {blo.z, blo.w};
        v2f b2 = {bhi.x, bhi.y}, b3 = {bhi.z, bhi.w};
        c = __builtin_amdgcn_wmma_f32_16x16x4_f32(false, a0, false, b0, (short)0, c, false, false);
        c = __builtin_amdgcn_wmma_f32_16x16x4_f32(false, a1, false, b1, (short)0, c, false, false);
        c = __builtin_amdgcn_wmma_f32_16x16x4_f32(false, a2, false, b2, (short)0, c, false, false);
        c = __builtin_amdgcn_wmma_f32_16x16x4_f32(false, a3, false, b3, (short)0, c, false, false);
    }

    // C/D layout: lanes 0-15 hold N=lane with VGPR j = row M=j (M=0..7);
    // lanes 16-31 hold M=8..15 (padding) -> discarded.
    if (lane < 16) {
        const int col = n0 + lane;
        const float bv = bias ? bias[col] : 0.f;
        #pragma unroll
        for (int m = 0; m < 8; ++m)
            out[(size_t)m * N + col] = c[m] + bv;
    }
}

// ---------------------------------------------------------------------------
// RoPE in-place on the q and k slices of the qkv workspace (pos = KV-1).
// ---------------------------------------------------------------------------
__global__ void rope_kernel(float* __restrict__ qkv,
                            const float* __restrict__ cos_t,
                            const float* __restrict__ sin_t)
{
    int t = blockIdx.x * blockDim.x + threadIdx.x;   // BATCH*H*(D/2) = 16384
    if (t >= BATCH * H * (D / 2)) return;
    const int b   = t / (H * (D / 2));
    const int rem = t % (H * (D / 2));
    const int h   = rem / (D / 2);
    const int i   = rem % (D / 2);
    const float c = cos_t[b * (D / 2) + i];
    const float s = sin_t[b * (D / 2) + i];

    float* q = qkv + (size_t)b * QKV3 + h * D;
    float x1 = q[i], x2 = q[i + D / 2];
    q[i]         = x1 * c - x2 * s;
    q[i + D / 2] = x1 * s + x2 * c;

    float* k = qkv + (size_t)b * QKV3 + HID + h * D;
    x1 = k[i]; x2 = k[i + D / 2];
    k[i]         = x1 * c - x2 * s;
    k[i + D / 2] = x1 * s + x2 * c;
}

// ---------------------------------------------------------------------------
// Decode attention: one workgroup per (b,h); 8 waves split the 2048 positions
// (online softmax per wave, lane-parallel over d), LDS merge at the end.
// The freshly written cache slot (slots[b]) is served from the roped k/v in
// the workspace, since the input caches are immutable.
// ---------------------------------------------------------------------------
__global__ __launch_bounds__(256) void attn_kernel(
    const float* __restrict__ qkv,       // post-rope, in ws
    const float* __restrict__ k_cache,
    const float* __restrict__ v_cache,
    const int*   __restrict__ block_tables,
    const int*   __restrict__ slots,
    const int*   __restrict__ input_lengths,
    float* __restrict__ attn)            // BATCH x HID
{
    const int bh   = blockIdx.x;
    const int b    = bh / H;
    const int h    = bh % H;
    const int lane = threadIdx.x & 31;
    const int wave = threadIdx.x >> 5;   // 0..7
    const int len  = input_lengths[b];
    const int myslot = slots[b];

    const float* __restrict__ qp   = qkv + (size_t)b * QKV3 + h * D;
    const float* __restrict__ knew = qkv + (size_t)b * QKV3 + HID + h * D;
    const float* __restrict__ vnew = qkv + (size_t)b * QKV3 + 2 * HID + h * D;
    const int*   __restrict__ bt   = block_tables + b * (KVLEN / BS);

    const float4 qf = *(const float4*)(qp + lane * 4);

    float  m = -3.0e38f, l = 0.f;
    float4 acc = make_float4(0.f, 0.f, 0.f, 0.f);

    const int per = KVLEN / 8;
    const int s0  = wave * per;
    const int ns  = min(s0 + per, len);

    #pragma unroll 2
    for (int s = s0; s < ns; ++s) {
        // keep the HBM stream ahead of the dependent softmax chain
        if (s + 8 < ns) {
            const int pidx = bt[(s + 8) >> 4] * BS + ((s + 8) & (BS - 1));
            const size_t poff = (size_t)pidx * HID + h * D + lane * 4;
            __builtin_prefetch(k_cache + poff, 0, 0);
            __builtin_prefetch(v_cache + poff, 0, 0);
        }
        const int idx = bt[s >> 4] * BS + (s & (BS - 1));
        const float* kr;
        const float* vr;
        if (idx == myslot) { kr = knew; vr = vnew; }
        else {
            kr = k_cache + (size_t)idx * HID + h * D;
            vr = v_cache + (size_t)idx * HID + h * D;
        }
        const float4 kf = *(const float4*)(kr + lane * 4);
        const float4 vf = *(const float4*)(vr + lane * 4);
        float part = qf.x * kf.x + qf.y * kf.y + qf.z * kf.z + qf.w * kf.w;
        #pragma unroll
        for (int off = 16; off > 0; off >>= 1)
            part += __shfl_xor(part, off, 32);
        const float sc = part * SCALE;               // wave-uniform

        const float mn   = fmaxf(m, sc);
        const float corr = __expf(m - mn);
        const float p    = __expf(sc - mn);
        acc.x = acc.x * corr + p * vf.x;
        acc.y = acc.y * corr + p * vf.y;
        acc.z = acc.z * corr + p * vf.z;
        acc.w = acc.w * corr + p * vf.w;
        l = l * corr + p;
        m = mn;
    }

    __shared__ float sm_m[8];
    __shared__ float sm_l[8];
    __shared__ float sm_acc[8][D];
    if (lane == 0) { sm_m[wave] = m; sm_l[wave] = l; }
    *(float4*)&sm_acc[wave][lane * 4] = acc;
    __syncthreads();

    if (wave == 0) {
        float mt = -3.0e38f;
        #pragma unroll
        for (int w = 0; w < 8; ++w) mt = fmaxf(mt, sm_m[w]);
        float lt = 0.f;
        float4 o = make_float4(0.f, 0.f, 0.f, 0.f);
        #pragma unroll
        for (int w = 0; w < 8; ++w) {
            const float f = __expf(sm_m[w] - mt);
            lt += f * sm_l[w];
            const float4 aw = *(const float4*)&sm_acc[w][lane * 4];
            o.x += f * aw.x; o.y += f * aw.y; o.z += f * aw.z; o.w += f * aw.w;
        }
        const float inv = 1.f / lt;
        float4 r = make_float4(o.x * inv, o.y * inv, o.z * inv, o.w * inv);
        *(float4*)(attn + (size_t)b * HID + h * D + lane * 4) = r;
    }
}

// ---------------------------------------------------------------------------
extern "C" void kernel_launch(void* const* d_in, const int* in_sizes, int n_in,
                              void* d_out, int out_size, void* d_ws, size_t ws_size,
                              hipStream_t stream)
{
    const float* hidden   = (const float*)d_in[0];
    const float* cos_t    = (const float*)d_in[1];
    const float* sin_t    = (const float*)d_in[2];
    const float* w_qkv    = (const float*)d_in[3];
    const float* b_qkv    = (const float*)d_in[4];
    const float* w_o      = (const float*)d_in[5];
    const float* k_cache  = (const float*)d_in[6];
    const float* v_cache  = (const float*)d_in[7];
    const int*   blk_tbl  = (const int*)d_in[8];
    const int*   slots    = (const int*)d_in[9];
    const int*   in_lens  = (const int*)d_in[10];
    (void)in_sizes; (void)n_in; (void)out_size; (void)ws_size;

    float* qkv  = (float*)d_ws;                  // BATCH*QKV3 floats
    float* attn = qkv + (size_t)BATCH * QKV3;    // BATCH*HID  floats
    float* out  = (float*)d_out;

    // 1) qkv = hidden @ w_qkv^T + b_qkv        (768 N-tiles / 8 waves)
    gemm_aw_t<<<96, 256, 0, stream>>>(hidden, w_qkv, b_qkv, qkv, QKV3, HID);
    // 2) RoPE in-place on q,k slices
    rope_kernel<<<64, 256, 0, stream>>>(qkv, cos_t, sin_t);
    // 3) attention
    attn_kernel<<<BATCH * H, 256, 0, stream>>>(qkv, k_cache, v_cache,
                                               blk_tbl, slots, in_lens, attn);
    // 4) out = attn @ w_o^T                    (256 N-tiles / 8 waves)
    gemm_aw_t<<<32, 256, 0, stream>>>(attn, w_o, nullptr, out, HID, HID);
}